// GCN_Module_39170101740054
// MI455X (gfx1250) — compile-verified
//
#include <hip/hip_runtime.h>

// MI455X / gfx1250, wave32. fp32 WMMA path: V_WMMA_F32_16X16X4_F32,
// GLOBAL_LOAD_ASYNC_TO_LDS_B128 double-buffered staging (ASYNCcnt).
// Both A and B are staged so each lane's WMMA operand pair is one
// contiguous 8B LDS load (ds_load_b64), no register shuffling.
typedef float v2f __attribute__((ext_vector_type(2)));
typedef float v8f __attribute__((ext_vector_type(8)));

#define BB 2
#define NN 10000
#define EE 150000
#define CC 256
#define K1 259          // C + 3
#define SW1 260         // Wt_e row stride (k-major), 1040B rows (16B mult)
#define SW2 256         // Wt_u row stride, 1024B rows
#define EPSBN 1e-5f

#define G1 512          // persistent workgroups, edge GEMM
#define G2 256          // persistent workgroups, node GEMM
#define TILES_PER_BATCH ((EE + 63) / 64)   // 2344
#define TILES1 (BB * TILES_PER_BATCH)      // 4688
#define JOBS1 (TILES1 * 2)                 // x2 channel halves
#define ROWS2 (BB * NN)                    // 20000
#define TILES2 ((ROWS2 + 63) / 64)         // 313
#define JOBS2 (TILES2 * 2)

#define INITKEY 0x007FFFFFu                // kmap(-inf)

// Monotone float -> uint map so atomicMax(u32) == float max.
__device__ __forceinline__ unsigned kmap(float x) {
  unsigned i = __float_as_uint(x);
  return (i & 0x80000000u) ? ~i : (i | 0x80000000u);
}
__device__ __forceinline__ float kunmap(unsigned k) {
  unsigned i = (k & 0x80000000u) ? (k ^ 0x80000000u) : ~k;
  return __uint_as_float(i);
}

// gfx1250 async copy: 16B global -> LDS, tracked by ASYNCcnt.
__device__ __forceinline__ void async_ld16(const void* g, void* lds) {
  asm volatile("global_load_async_to_lds_b128 %0, %1, off"
               :
               : "v"((unsigned)(size_t)lds),
                 "v"((unsigned long long)(size_t)g)
               : "memory");
}
#define ASYNC_WAIT(n) \
  asm volatile("s_wait_asynccnt %0" ::"n"(n) : "memory")

// One K-chunk of 8 WMMA steps (32 K). A: [m][k] stride 36. B: [n][k]
// stride 36 (k-major) -> both operands are single ds_load_b64 per lane.
__device__ __forceinline__ void mma_chunk(const float* __restrict__ Ab,
                                          const float* __restrict__ Bb,
                                          int aRow, int bRow0, int laneHi,
                                          v8f acc[4]) {
#pragma unroll
  for (int s = 0; s < 8; ++s) {
    const int k0 = s * 4 + 2 * laneHi;
    const v2f a = *(const v2f*)&Ab[aRow + k0];
#pragma unroll
    for (int j = 0; j < 4; ++j) {
      const v2f bv = *(const v2f*)&Bb[bRow0 + j * (16 * 36) + k0];
      acc[j] = __builtin_amdgcn_wmma_f32_16x16x4_f32(
          false, a, false, bv, (short)0, acc[j], false, false);
    }
  }
}

__device__ __forceinline__ void mma_step(const float* __restrict__ Ab,
                                         const float* __restrict__ Bb,
                                         int aRow, int bRow0, int laneHi,
                                         v8f acc[4]) {
  const int k0 = 2 * laneHi;
  const v2f a = *(const v2f*)&Ab[aRow + k0];
#pragma unroll
  for (int j = 0; j < 4; ++j) {
    const v2f bv = *(const v2f*)&Bb[bRow0 + j * (16 * 36) + k0];
    acc[j] = __builtin_amdgcn_wmma_f32_16x16x4_f32(
        false, a, false, bv, (short)0, acc[j], false, false);
  }
}

// ---------------- features [B,C,N] -> vf [B,N,C] (LDS-tiled transpose) ----
__global__ __launch_bounds__(256) void transpose_feat(
    const float* __restrict__ f, float* __restrict__ vf) {
  __shared__ float t[32][33];
  const int b = blockIdx.z, c0 = blockIdx.y * 32, n0 = blockIdx.x * 32;
  const int j = threadIdx.x & 31, i8 = threadIdx.x >> 5;
#pragma unroll
  for (int ii = 0; ii < 4; ++ii) {
    const int i = ii * 8 + i8;
    const int n = n0 + j;
    t[i][j] = (n < NN) ? f[((size_t)b * CC + c0 + i) * NN + n] : 0.f;
  }
  __syncthreads();
#pragma unroll
  for (int ii = 0; ii < 4; ++ii) {
    const int nL = ii * 8 + i8;
    const int n = n0 + nL;
    if (n < NN) vf[((size_t)b * NN + n) * CC + c0 + j] = t[j][nL];
  }
}

// ------- W [R][256] (row-major) -> Wt [256][S] (k-major, zero padded) -----
__global__ __launch_bounds__(256) void transpose_w(
    const float* __restrict__ src, float* __restrict__ dst, int R, int S) {
  __shared__ float t[32][33];
  const int k0 = blockIdx.x * 32, n0 = blockIdx.y * 32;
  const int j = threadIdx.x & 31, i8 = threadIdx.x >> 5;
#pragma unroll
  for (int ii = 0; ii < 4; ++ii) {
    const int i = ii * 8 + i8;
    t[i][j] = (k0 + i < R) ? src[(size_t)(k0 + i) * CC + n0 + j] : 0.f;
  }
  __syncthreads();
#pragma unroll
  for (int ii = 0; ii < 4; ++ii) {
    const int nL = ii * 8 + i8;
    if (k0 + j < S) dst[(size_t)(n0 + nL) * S + k0 + j] = t[j][nL];
  }
}

// ---------------- init agg keys to mapped(-inf) ---------------------------
__global__ void init_agg(unsigned* __restrict__ agg) {
  const size_t i = (size_t)blockIdx.x * 256 + threadIdx.x;
  if (i < (size_t)BB * NN * CC) agg[i] = INITKEY;
}

// ---------------- edge GEMM + ReLU + scatter-max + BN1 partial stats ------
// Tile: 64 edges x 128 out-channels. 8 waves: wave w -> mSub=w&3,
// nSubs = (w>>2)*4 + {0..3}. K: 8 async double-buffered chunks of 32
// (features) + 1 scalar tail chunk of 4 (xyz diff + zero pad).
__global__ __launch_bounds__(256) void edge_gemm(
    const float* __restrict__ vf, const float* __restrict__ xyz,
    const int* __restrict__ edges, const float* __restrict__ Wt,
    const float* __restrict__ be, unsigned* __restrict__ agg,
    float* __restrict__ ps, float* __restrict__ pq) {
  __shared__ float As[2][64 * 36];    // [64 m][32 k] pad->36 (144B rows)
  __shared__ float Bt[2][128 * 36];   // [128 n][32 k] pad->36 (144B rows)
  __shared__ int sSrc[64], sDst[64];
  __shared__ float sDiff[64][3];

  const int tid = threadIdx.x;
  const int lane = tid & 31, wv = tid >> 5;
  const int laneLo = lane & 15, laneHi = lane >> 4;
  const int mSub = wv & 3;
  const int nBase = (wv >> 2) * 4;
  const int aRow = (mSub * 16 + laneLo) * 36;
  const int bRow0 = (nBase * 16 + laneLo) * 36;

  // async A mapping: thread covers 2 of 512 16B segments (64 rows x 8 seg)
  const int am0 = tid >> 3, am1 = (tid + 256) >> 3;
  const int asg = (tid & 7) * 4;
  // async B mapping: thread covers 4 of 1024 16B segments (128 rows x 8 seg)
  const int bn = tid >> 3;
  const int bsg = (tid & 7) * 4;

  float accS0[4] = {0, 0, 0, 0}, accQ0[4] = {0, 0, 0, 0};
  float accS1[4] = {0, 0, 0, 0}, accQ1[4] = {0, 0, 0, 0};

  for (int job = blockIdx.x; job < JOBS1; job += G1) {
    const int tile = job >> 1, half = job & 1;
    const int b = tile / TILES_PER_BATCH;
    const int e0 = (tile - b * TILES_PER_BATCH) * 64;
    const int valid = min(64, EE - e0);
    const size_t vbase = (size_t)b * NN;
    const float* WtH = Wt + (size_t)(half * 128) * SW1;   // k-major rows

    __syncthreads();
    if (tid < 64) {
      int s = 0, d = 0;
      if (tid < valid) {
        s = edges[(size_t)(b * 2) * EE + e0 + tid];
        d = edges[(size_t)(b * 2 + 1) * EE + e0 + tid];
      }
      sSrc[tid] = s; sDst[tid] = d;
      const float* xs = xyz + (vbase + s) * 3;
      const float* xd = xyz + (vbase + d) * 3;
      sDiff[tid][0] = xs[0] - xd[0];
      sDiff[tid][1] = xs[1] - xd[1];
      sDiff[tid][2] = xs[2] - xd[2];
    }
    __syncthreads();

    v8f acc[4] = {};
    if (valid == 64) {
      // -------- fast path: async double-buffered staging, K = 0..255 -----
      {
        async_ld16(vf + (vbase + sSrc[am0]) * CC + asg,
                   &As[0][am0 * 36 + asg]);
        async_ld16(vf + (vbase + sSrc[am1]) * CC + asg,
                   &As[0][am1 * 36 + asg]);
#pragma unroll
        for (int i = 0; i < 4; ++i) {
          const int n = bn + i * 32;
          async_ld16(WtH + (size_t)n * SW1 + bsg, &Bt[0][n * 36 + bsg]);
        }
      }
      for (int c = 0; c < 8; ++c) {
        if (c < 7) {
          const int k0c = (c + 1) * 32;
          float* An = As[(c + 1) & 1];
          float* Bn = Bt[(c + 1) & 1];
          async_ld16(vf + (vbase + sSrc[am0]) * CC + k0c + asg,
                     &An[am0 * 36 + asg]);
          async_ld16(vf + (vbase + sSrc[am1]) * CC + k0c + asg,
                     &An[am1 * 36 + asg]);
#pragma unroll
          for (int i = 0; i < 4; ++i) {
            const int n = bn + i * 32;
            async_ld16(WtH + (size_t)n * SW1 + k0c + bsg,
                       &Bn[n * 36 + bsg]);
          }
          ASYNC_WAIT(6);   // only next chunk's 6 loads may remain in flight
        } else {
          ASYNC_WAIT(0);
        }
        __syncthreads();
        mma_chunk(As[c & 1], Bt[c & 1], aRow, bRow0, laneHi, acc);
        __syncthreads();
      }
    } else {
      // -------- slow path (tail tiles only): scalar staging --------------
      for (int k0c = 0; k0c < 256; k0c += 32) {
        for (int idx = tid; idx < 64 * 32; idx += 256) {
          const int m = idx >> 5, kk = idx & 31;
          As[0][m * 36 + kk] =
              (m < valid) ? vf[(vbase + sSrc[m]) * CC + k0c + kk] : 0.f;
        }
        for (int idx = tid; idx < 128 * 32; idx += 256) {
          const int n = idx >> 5, kk = idx & 31;
          Bt[0][n * 36 + kk] = WtH[(size_t)n * SW1 + k0c + kk];
        }
        __syncthreads();
        mma_chunk(As[0], Bt[0], aRow, bRow0, laneHi, acc);
        __syncthreads();
      }
    }
    // -------- K tail: k = 256..259 (xyz diff cols; Wt row 259 is zero) ---
    for (int idx = tid; idx < 64 * 4; idx += 256) {
      const int m = idx >> 2, kk = idx & 3;
      As[0][m * 36 + kk] = (m < valid && kk < 3) ? sDiff[m][kk] : 0.f;
    }
    for (int idx = tid; idx < 128 * 4; idx += 256) {
      const int n = idx >> 2, kk = idx & 3;
      Bt[0][n * 36 + kk] = WtH[(size_t)n * SW1 + 256 + kk];
    }
    __syncthreads();
    mma_step(As[0], Bt[0], aRow, bRow0, laneHi, acc);

    // Epilogue: +bias, ReLU, scatter-max raw h (BN1 affine has a>0, so
    // max commutes with BN1), deterministic per-wave BN1 partial sums.
    if (valid == 64) {
#pragma unroll
      for (int j = 0; j < 4; ++j) {
        const int ch = half * 128 + (nBase + j) * 16 + laneLo;
        const float bias = be[ch];
        float s = 0.f, q = 0.f;
#pragma unroll
        for (int r = 0; r < 8; ++r) {
          const int m = mSub * 16 + r + 8 * laneHi;
          float h = acc[j][r] + bias;
          h = h > 0.f ? h : 0.f;
          atomicMax(&agg[(vbase + sDst[m]) * CC + ch], kmap(h));
          s += h; q += h * h;
        }
        s += __shfl_down(s, 16);
        q += __shfl_down(q, 16);
        if (half == 0) { accS0[j] += s; accQ0[j] += q; }
        else           { accS1[j] += s; accQ1[j] += q; }
      }
    } else {
#pragma unroll
      for (int j = 0; j < 4; ++j) {
        const int ch = half * 128 + (nBase + j) * 16 + laneLo;
        const float bias = be[ch];
        float s = 0.f, q = 0.f;
#pragma unroll
        for (int r = 0; r < 8; ++r) {
          const int m = mSub * 16 + r + 8 * laneHi;
          float h = acc[j][r] + bias;
          h = h > 0.f ? h : 0.f;
          if (m < valid) {
            atomicMax(&agg[(vbase + sDst[m]) * CC + ch], kmap(h));
            s += h; q += h * h;
          }
        }
        s += __shfl_down(s, 16);
        q += __shfl_down(q, 16);
        if (half == 0) { accS0[j] += s; accQ0[j] += q; }
        else           { accS1[j] += s; accQ1[j] += q; }
      }
    }
  }
  if (lane < 16) {
#pragma unroll
    for (int j = 0; j < 4; ++j) {
      const int base = ((size_t)blockIdx.x * 8 + wv) * 128 + j * 16 + lane;
      ps[base]      = accS0[j]; pq[base]      = accQ0[j];
      ps[base + 64] = accS1[j]; pq[base + 64] = accQ1[j];
    }
  }
}

// ---------------- BN stat reduction -> per-channel affine (a, c) ----------
__global__ void bn_reduce(const float* __restrict__ ps,
                          const float* __restrict__ pq, int G, float invCount,
                          const float* __restrict__ g,
                          const float* __restrict__ beta,
                          float* __restrict__ a, float* __restrict__ c) {
  const int ch = threadIdx.x;                // 256 threads, 1 block
  const int half = ch >> 7, r = ch & 127, nSub = r >> 4, lane = r & 15;
  const int jj = nSub & 3, vb = (nSub < 4) ? 0 : 4;
  float s = 0.f, q = 0.f;
  for (int w = 0; w < G; ++w)
    for (int v = vb; v < vb + 4; ++v) {
      const int idx = (w * 8 + v) * 128 + half * 64 + jj * 16 + lane;
      s += ps[idx]; q += pq[idx];
    }
  const float mean = s * invCount;
  const float var = fmaxf(q * invCount - mean * mean, 0.f);
  const float aa = g[ch] * rsqrtf(var + EPSBN);
  a[ch] = aa;
  c[ch] = beta[ch] - mean * aa;
}

// ---------------- node GEMM: BN1(decode agg) @ Wu, ReLU, BN2 stats --------
__global__ __launch_bounds__(256) void node_gemm(
    const unsigned* __restrict__ agg, const float* __restrict__ Wt,
    const float* __restrict__ bu, const float* __restrict__ ae,
    const float* __restrict__ ce, float* __restrict__ y2,
    float* __restrict__ ps, float* __restrict__ pq) {
  __shared__ float As[64 * 36];
  __shared__ float Bt[128 * 36];
  __shared__ float sA[256], sC[256];

  const int tid = threadIdx.x;
  const int lane = tid & 31, wv = tid >> 5;
  const int laneLo = lane & 15, laneHi = lane >> 4;
  const int mSub = wv & 3;
  const int nBase = (wv >> 2) * 4;
  const int aRow = (mSub * 16 + laneLo) * 36;
  const int bRow0 = (nBase * 16 + laneLo) * 36;
  const int bn = tid >> 3, bsg = (tid & 7) * 4;

  sA[tid] = ae[tid];
  sC[tid] = ce[tid];

  float accS0[4] = {0, 0, 0, 0}, accQ0[4] = {0, 0, 0, 0};
  float accS1[4] = {0, 0, 0, 0}, accQ1[4] = {0, 0, 0, 0};

  for (int job = blockIdx.x; job < JOBS2; job += G2) {
    const int tile = job >> 1, half = job & 1;
    const int r0 = tile * 64;
    const float* WtH = Wt + (size_t)(half * 128) * SW2;

    v8f acc[4] = {};
    for (int k0c = 0; k0c < CC; k0c += 32) {
      __syncthreads();
      // async B staging overlaps the scalar A decode below
#pragma unroll
      for (int i = 0; i < 4; ++i) {
        const int n = bn + i * 32;
        async_ld16(WtH + (size_t)n * SW2 + k0c + bsg, &Bt[n * 36 + bsg]);
      }
      for (int idx = tid; idx < 64 * 32; idx += 256) {
        const int m = idx >> 5, kk = idx & 31;
        const int k = k0c + kk;
        const int row = r0 + m;
        float v = 0.f;
        if (row < ROWS2) {
          const unsigned key = agg[(size_t)row * CC + k];
          if (key != INITKEY) v = fmaf(sA[k], kunmap(key), sC[k]);
          // empty segment: reference fills 0 AFTER BN -> v stays 0
        }
        As[m * 36 + kk] = v;
      }
      ASYNC_WAIT(0);
      __syncthreads();
      mma_chunk(As, Bt, aRow, bRow0, laneHi, acc);
    }
#pragma unroll
    for (int j = 0; j < 4; ++j) {
      const int ch = half * 128 + (nBase + j) * 16 + laneLo;
      const float bias = bu[ch];
      float s = 0.f, q = 0.f;
#pragma unroll
      for (int r = 0; r < 8; ++r) {
        const int row = r0 + mSub * 16 + r + 8 * laneHi;
        float h = acc[j][r] + bias;
        h = h > 0.f ? h : 0.f;
        if (row < ROWS2) {
          y2[(size_t)row * CC + ch] = h;
          s += h; q += h * h;
        }
      }
      s += __shfl_down(s, 16);
      q += __shfl_down(q, 16);
      if (half == 0) { accS0[j] += s; accQ0[j] += q; }
      else           { accS1[j] += s; accQ1[j] += q; }
    }
  }
  if (lane < 16) {
#pragma unroll
    for (int j = 0; j < 4; ++j) {
      const int base = ((size_t)blockIdx.x * 8 + wv) * 128 + j * 16 + lane;
      ps[base]      = accS0[j]; pq[base]      = accQ0[j];
      ps[base + 64] = accS1[j]; pq[base + 64] = accQ1[j];
    }
  }
}

// ---------------- finalize: out[b,c,n] = BN2(y2[b,n,c]) + feat[b,c,n] -----
__global__ __launch_bounds__(256) void finalize(
    const float* __restrict__ y2, const float* __restrict__ au,
    const float* __restrict__ cu, const float* __restrict__ feat,
    float* __restrict__ out) {
  __shared__ float t[32 * 257];
  const int b = blockIdx.y;
  const int n0 = blockIdx.x * 32;
  const int tid = threadIdx.x;
  for (int idx = tid; idx < 32 * 256; idx += 256) {
    const int nL = idx >> 8, c = idx & 255;
    const int row = b * NN + n0 + nL;
    t[nL * 257 + c] = (n0 + nL < NN) ? y2[(size_t)row * CC + c] : 0.f;
  }
  __syncthreads();
  const int nL = tid & 31, cBase = tid >> 5;
  for (int it = 0; it < 32; ++it) {
    const int c = it * 8 + cBase;
    const int n = n0 + nL;
    if (n < NN) {
      const size_t gi = ((size_t)b * CC + c) * NN + n;
      out[gi] = fmaf(au[c], t[nL * 257 + c], cu[c]) + feat[gi];
    }
  }
}

extern "C" void kernel_launch(void* const* d_in, const int* in_sizes, int n_in,
                              void* d_out, int out_size, void* d_ws,
                              size_t ws_size, hipStream_t stream) {
  const float* xyz  = (const float*)d_in[0];
  const float* feat = (const float*)d_in[1];
  const int*   edges= (const int*)d_in[2];
  const float* We   = (const float*)d_in[3];
  const float* be   = (const float*)d_in[4];
  const float* ge   = (const float*)d_in[5];
  const float* bte  = (const float*)d_in[6];
  const float* Wu   = (const float*)d_in[7];
  const float* bu   = (const float*)d_in[8];
  const float* gu   = (const float*)d_in[9];
  const float* btu  = (const float*)d_in[10];
  float* out = (float*)d_out;

  float* ws = (float*)d_ws;
  size_t o = 0;
  float*    vf  = ws + o; o += (size_t)BB * NN * CC;
  unsigned* agg = (unsigned*)(ws + o); o += (size_t)BB * NN * CC;
  float*    y2  = ws + o; o += (size_t)BB * NN * CC;
  float*    Wt1 = ws + o; o += (size_t)CC * SW1;   // We^T, k-major
  float*    Wt2 = ws + o; o += (size_t)CC * SW2;   // Wu^T, k-major
  float*    ps1 = ws + o; o += (size_t)G1 * 8 * 128;
  float*    pq1 = ws + o; o += (size_t)G1 * 8 * 128;
  float*    ps2 = ws + o; o += (size_t)G2 * 8 * 128;
  float*    pq2 = ws + o; o += (size_t)G2 * 8 * 128;
  float*    ae  = ws + o; o += 256;
  float*    ce  = ws + o; o += 256;
  float*    au  = ws + o; o += 256;
  float*    cu  = ws + o; o += 256;

  const dim3 bt(256);
  transpose_feat<<<dim3((NN + 31) / 32, CC / 32, BB), bt, 0, stream>>>(feat, vf);
  transpose_w<<<dim3((SW1 + 31) / 32, CC / 32), bt, 0, stream>>>(We, Wt1, K1, SW1);
  transpose_w<<<dim3(SW2 / 32, CC / 32), bt, 0, stream>>>(Wu, Wt2, CC, SW2);
  init_agg<<<dim3((BB * NN * CC + 255) / 256), bt, 0, stream>>>(agg);
  edge_gemm<<<dim3(G1), bt, 0, stream>>>(vf, xyz, edges, Wt1, be, agg, ps1, pq1);
  bn_reduce<<<dim3(1), bt, 0, stream>>>(ps1, pq1, G1, 1.0f / (float)(BB * EE),
                                        ge, bte, ae, ce);
  node_gemm<<<dim3(G2), bt, 0, stream>>>(agg, Wt2, bu, ae, ce, y2, ps2, pq2);
  bn_reduce<<<dim3(1), bt, 0, stream>>>(ps2, pq2, G2, 1.0f / (float)(BB * NN),
                                        gu, btu, au, cu);
  finalize<<<dim3((NN + 31) / 32, BB), bt, 0, stream>>>(y2, au, cu, feat, out);
}